// NodeAggregator_65068754534511
// MI455X (gfx1250) — compile-verified
//
#include <hip/hip_runtime.h>

typedef __attribute__((ext_vector_type(2))) float v2f;
typedef __attribute__((ext_vector_type(8))) float v8f;

#define IN_F    128
#define OUT_F   128
#define NNODES  50000
#define KNB     32
#define KDIM    256           // 2*IN_F
#define TILE_M  16
#define STRD    260           // padded row stride (words); 260 % 64 == 4 -> conflict-free WMMA reads
#define NTILES  (NNODES / TILE_M)   // 3125

__global__ __launch_bounds__(256)
void node_agg_wmma_f32(const float* __restrict__ v,
                       const float* __restrict__ nbr,
                       const float* __restrict__ W,
                       const float* __restrict__ bias,
                       float* __restrict__ out)
{
    __shared__ float sW[OUT_F * STRD];   // W[o][k], padded rows   (~133 KB)
    __shared__ float sA[TILE_M * STRD];  // A[i][k] = [32*v | sum_k nbr]  (~17 KB)

    const int t    = threadIdx.x;        // 0..255
    const int lane = t & 31;
    const int wv   = t >> 5;             // wave id 0..7
    const int hf   = lane >> 4;          // half-wave 0/1
    const int m    = lane & 15;          // row/col within 16
    const int ncol = wv * 16 + m;        // this wave's output column for lane

    // ---- Phase 0 (once): W (row-major) -> padded LDS, coalesced ----
    for (int o = 0; o < OUT_F; ++o) {
        sW[o * STRD + t] = W[o * KDIM + t];
    }
    const float bcol = 32.0f * bias[ncol];   // constant per lane across all tiles
    __syncthreads();

    const int irow = t >> 4;             // 0..15 : node row within tile
    const int dq   = (t & 15) * 4;       // 0..60 : feature quad offset

    for (int tile = blockIdx.x; tile < NTILES; tile += gridDim.x) {
        const int n0   = tile * TILE_M;
        const int node = n0 + irow;

        // ---- Phase 1: build A-tile = [32*v | sum_k neighbors] ----
        {
            const float* vp = v + (long)node * IN_F;
            const float4 va = *(const float4*)(vp + dq);
            const float4 vb = *(const float4*)(vp + dq + 64);

            float4 s0 = make_float4(0.f, 0.f, 0.f, 0.f);
            float4 s1 = make_float4(0.f, 0.f, 0.f, 0.f);
            #pragma unroll 4
            for (int k = 0; k < KNB; ++k) {
                const float* np = nbr + ((long)k * NNODES + node) * IN_F;
                const float4 x0 = *(const float4*)(np + dq);
                const float4 x1 = *(const float4*)(np + dq + 64);
                s0.x += x0.x; s0.y += x0.y; s0.z += x0.z; s0.w += x0.w;
                s1.x += x1.x; s1.y += x1.y; s1.z += x1.z; s1.w += x1.w;
            }

            float* ar = &sA[irow * STRD];
            *(float4*)(ar + dq)       = make_float4(32.f*va.x, 32.f*va.y, 32.f*va.z, 32.f*va.w);
            *(float4*)(ar + dq + 64)  = make_float4(32.f*vb.x, 32.f*vb.y, 32.f*vb.z, 32.f*vb.w);
            *(float4*)(ar + 128 + dq) = s0;
            *(float4*)(ar + 192 + dq) = s1;
        }
        __syncthreads();

        // ---- Phase 2: 16x16 output tile per wave via V_WMMA_F32_16X16X4_F32 ----
        v8f c = {0.f, 0.f, 0.f, 0.f, 0.f, 0.f, 0.f, 0.f};
        const float* aB = &sA[m * STRD];        // A row m
        const float* bB = &sW[ncol * STRD];     // W row ncol (acts as B column ncol)
        #pragma unroll 8
        for (int k0 = 0; k0 < KDIM; k0 += 4) {
            const int kk = k0 + 2 * hf;
            // A 16x4 layout: VGPR0 = A[m][kk], VGPR1 = A[m][kk+1]
            v2f av = { aB[kk], aB[kk + 1] };
            // B 4x16 layout: VGPR0 = B[kk][n], VGPR1 = B[kk+1][n]; B[k][o] = W[o][k]
            v2f bv = { bB[kk], bB[kk + 1] };
            c = __builtin_amdgcn_wmma_f32_16x16x4_f32(
                    /*neg_a=*/false, av, /*neg_b=*/false, bv,
                    /*c_mod=*/(short)0, c, /*reuse_a=*/false, /*reuse_b=*/false);
        }

        // ---- Epilogue: bias + store. D VGPR r holds row (r + 8*hf), col ncol ----
        #pragma unroll
        for (int r = 0; r < 8; ++r) {
            out[(long)(n0 + r + 8 * hf) * OUT_F + ncol] = c[r] + bcol;
        }
        __syncthreads();   // protect sA before next tile's Phase 1
    }
}

extern "C" void kernel_launch(void* const* d_in, const int* in_sizes, int n_in,
                              void* d_out, int out_size, void* d_ws, size_t ws_size,
                              hipStream_t stream) {
    const float* v    = (const float*)d_in[0];   // (50000, 128)
    const float* nbr  = (const float*)d_in[1];   // (32, 50000, 128)
    const float* W    = (const float*)d_in[2];   // (128, 256)
    const float* bias = (const float*)d_in[3];   // (128,)
    float* out        = (float*)d_out;           // (50000, 128)

    (void)in_sizes; (void)n_in; (void)out_size; (void)d_ws; (void)ws_size;

    node_agg_wmma_f32<<<dim3(1024), dim3(256), 0, stream>>>(v, nbr, W, bias, out);
}